// ZapBenchVolume_79001628442822
// MI455X (gfx1250) — compile-verified
//
#include <hip/hip_runtime.h>

// Quadrilinear (t,z,y,x) interpolation over frames[3][72][512][1024] (fp32).
// Pure random-gather, memory-bound kernel (AI ~ 0.03 FLOP/B): the only lever
// on MI455X is the data-movement path, not the ALU/matrix path.
//
// MI455X-specific choices:
//  - NT temporal hints (th:TH_LOAD_NT / store NT) on streaming coords/output
//    so the 192MB L2 is reserved for the 452MB frame volume (only reused data).
//  - x0/x0+1 gather pairs laid out so the backend merges them into b64 loads.
//  - 4 points per thread with explicit phase separation (addr math -> one
//    cluster of 32 independent gathers -> blend) to maximize outstanding
//    loads per wave32 (LOADcnt window), the latency-hiding lever for gathers.
//  - temporal-lerp-first factorization: 15 fma-lerps per point instead of 31.

typedef float v4f __attribute__((ext_vector_type(4)));

namespace {
constexpr int Tn  = 3;
constexpr int Zn  = 72;
constexpr int Yn  = 512;
constexpr int Xn  = 1024;
constexpr int YX  = Yn * Xn;        // 524288
constexpr int ZYX = Zn * YX;        // 37748736 elements per frame
constexpr int BLK = 256;            // 8 wave32 per block
constexpr int PPT = 4;              // points per thread
}

__device__ __forceinline__ float lerpf(float a, float b, float w) {
    return __builtin_fmaf(w, b - a, a);
}

struct PointCtx {
    const float* f0;   // frame t0 base
    const float* f1;   // frame t1 base
    int r00, r01, r10, r11;
    float wt, wz, wy, wx;
};

__device__ __forceinline__ PointCtx setup_point(const v4f c,
                                                const float* __restrict__ frames) {
    PointCtx p;
    // temporal axis
    const float ta = c.w * 3.0f;                 // t_abs in [0,3)
    int t0 = (int)ta;                            // floor (ta >= 0)
    t0 = min(t0, Tn - 1);
    const int t1 = min(t0 + 1, Tn - 1);
    p.wt = ta - (float)t0;

    // spatial axes (coords order: z, y, x)
    const float sz = c.x * (float)(Zn - 1);
    const float sy = c.y * (float)(Yn - 1);
    const float sx = c.z * (float)(Xn - 1);
    const int z0 = min(max((int)sz, 0), Zn - 2); // v_med3_i32
    const int y0 = min(max((int)sy, 0), Yn - 2);
    const int x0 = min(max((int)sx, 0), Xn - 2);
    // Recomputing weights after the clamp is exactly equivalent to the
    // reference's clip-after-floor semantics (w->1 at voxel dim-1 instead of
    // w->0 at the duplicated corner).
    p.wz = sz - (float)z0;
    p.wy = sy - (float)y0;
    p.wx = sx - (float)x0;

    p.r00 = z0 * YX + y0 * Xn + x0;              // (z0,y0) row base
    p.r01 = p.r00 + Xn;                          // (z0,y1)
    p.r10 = p.r00 + YX;                          // (z1,y0)
    p.r11 = p.r10 + Xn;                          // (z1,y1)
    p.f0 = frames + (size_t)t0 * (size_t)ZYX;
    p.f1 = frames + (size_t)t1 * (size_t)ZYX;
    return p;
}

__device__ __forceinline__ void gather_point(const PointCtx& p, float g[16]) {
    // 8 adjacent-pair gathers; backend merges each pair into global_load_b64.
    g[0]  = p.f0[p.r00]; g[1]  = p.f0[p.r00 + 1];
    g[2]  = p.f0[p.r01]; g[3]  = p.f0[p.r01 + 1];
    g[4]  = p.f0[p.r10]; g[5]  = p.f0[p.r10 + 1];
    g[6]  = p.f0[p.r11]; g[7]  = p.f0[p.r11 + 1];
    g[8]  = p.f1[p.r00]; g[9]  = p.f1[p.r00 + 1];
    g[10] = p.f1[p.r01]; g[11] = p.f1[p.r01 + 1];
    g[12] = p.f1[p.r10]; g[13] = p.f1[p.r10 + 1];
    g[14] = p.f1[p.r11]; g[15] = p.f1[p.r11 + 1];
}

__device__ __forceinline__ float blend_point(const PointCtx& p, const float g[16]) {
    // temporal lerp first (multilinear => commutes), then one trilerp
    const float c000 = lerpf(g[0], g[8],  p.wt);
    const float c001 = lerpf(g[1], g[9],  p.wt);
    const float c010 = lerpf(g[2], g[10], p.wt);
    const float c011 = lerpf(g[3], g[11], p.wt);
    const float c100 = lerpf(g[4], g[12], p.wt);
    const float c101 = lerpf(g[5], g[13], p.wt);
    const float c110 = lerpf(g[6], g[14], p.wt);
    const float c111 = lerpf(g[7], g[15], p.wt);

    const float c00 = lerpf(c000, c001, p.wx);
    const float c01 = lerpf(c010, c011, p.wx);
    const float c10 = lerpf(c100, c101, p.wx);
    const float c11 = lerpf(c110, c111, p.wx);
    const float c0  = lerpf(c00, c01, p.wy);
    const float c1  = lerpf(c10, c11, p.wy);
    return lerpf(c0, c1, p.wz);
}

__global__ __launch_bounds__(BLK)
void ZapBenchVolume_quadlerp_kernel(const float* __restrict__ coords,
                                    const float* __restrict__ frames,
                                    float* __restrict__ out,
                                    int n) {
    const v4f* __restrict__ c4 = reinterpret_cast<const v4f*>(coords);
    const int base = blockIdx.x * (BLK * PPT) + threadIdx.x;

    if (base + (PPT - 1) * BLK < n) {
        // ---- fast path: straight-line, phase-separated for max MLP --------
        v4f c[PPT];
#pragma unroll
        for (int p = 0; p < PPT; ++p)
            c[p] = __builtin_nontemporal_load(c4 + (base + p * BLK));

        PointCtx pc[PPT];
#pragma unroll
        for (int p = 0; p < PPT; ++p)
            pc[p] = setup_point(c[p], frames);

        float g[PPT][16];
#pragma unroll
        for (int p = 0; p < PPT; ++p)        // one cluster of 32 b64 gathers
            gather_point(pc[p], g[p]);

#pragma unroll
        for (int p = 0; p < PPT; ++p) {
            const float r = blend_point(pc[p], g[p]);
            __builtin_nontemporal_store(r, out + (base + p * BLK));
        }
    } else {
        // ---- tail path: per-point guard -----------------------------------
#pragma unroll
        for (int p = 0; p < PPT; ++p) {
            const int i = base + p * BLK;
            if (i >= n) continue;
            const v4f c = __builtin_nontemporal_load(c4 + i);
            const PointCtx pc = setup_point(c, frames);
            float g[16];
            gather_point(pc, g);
            const float r = blend_point(pc, g);
            __builtin_nontemporal_store(r, out + i);
        }
    }
}

extern "C" void kernel_launch(void* const* d_in, const int* in_sizes, int n_in,
                              void* d_out, int out_size, void* d_ws, size_t ws_size,
                              hipStream_t stream) {
    (void)n_in; (void)d_ws; (void)ws_size;
    const float* coords = (const float*)d_in[0];   // (N,4) fp32
    const float* frames = (const float*)d_in[1];   // (3,72,512,1024) fp32
    float* out = (float*)d_out;                    // (N,1) fp32

    const int n = out_size;                        // N points
    const int pts_per_block = BLK * PPT;
    const int grid = (n + pts_per_block - 1) / pts_per_block;

    ZapBenchVolume_quadlerp_kernel<<<grid, BLK, 0, stream>>>(coords, frames, out, n);
}